// OneUnkMaterialSparseStack_NK_38448547234792
// MI455X (gfx1250) — compile-verified
//
#include <hip/hip_runtime.h>
#include <hip/hip_bf16.h>

#define TWO_PI 6.283185307179586f

// Problem sizes (match reference constants)
#define KW      1024   // wavelengths
#define KA      64     // angles
#define KNPTS   350
#define KNSMOOTH 331
#define KKSIZE  20
#define KMAXD   400
#define KLABOVE 20
#define KLBELOW 19
#define KNFIXED 4
#define KNINT   40     // interior layers = 20 + 1 + 19
#define KNMATD  42     // material_distribution length
#define KNMAT   5      // 4 fixed + 1 unknown
#define KBLOCK  256
#define KGRID   ((KW * KA) / KBLOCK)   // 256

struct cplx { float x, y; };

__device__ __forceinline__ cplx cmul(cplx a, cplx b) {
  return { fmaf(a.x, b.x, -a.y * b.y), fmaf(a.x, b.y, a.y * b.x) };
}
__device__ __forceinline__ cplx cmadd(cplx a, cplx b, cplx c) {  // a*b + c
  return { fmaf(a.x, b.x, fmaf(-a.y, b.y, c.x)),
           fmaf(a.x, b.y, fmaf( a.y, b.x, c.y)) };
}
__device__ __forceinline__ cplx cadd(cplx a, cplx b) { return { a.x + b.x, a.y + b.y }; }
__device__ __forceinline__ cplx csub(cplx a, cplx b) { return { a.x - b.x, a.y - b.y }; }
__device__ __forceinline__ cplx cscale(cplx a, float s) { return { a.x * s, a.y * s }; }
__device__ __forceinline__ cplx crecip(cplx a) {
  float inv = 1.0f / fmaf(a.x, a.x, a.y * a.y);
  return { a.x * inv, -a.y * inv };
}
__device__ __forceinline__ cplx cdiv(cplx a, cplx b) { return cmul(a, crecip(b)); }
__device__ __forceinline__ cplx csqrtc(cplx z) {   // principal branch, matches jnp.sqrt(complex64)
  float r  = sqrtf(fmaf(z.x, z.x, z.y * z.y));
  float re = sqrtf(0.5f * fmaxf(r + z.x, 0.0f));
  float im = sqrtf(0.5f * fmaxf(r - z.x, 0.0f));
  if (z.y < 0.0f) im = -im;
  return { re, im };
}
// Select among 5 register-resident complex values by runtime index (cndmask chain,
// avoids scratch from dynamically indexed local arrays).
__device__ __forceinline__ cplx sel5(cplx v0, cplx v1, cplx v2, cplx v3, cplx v4, int m) {
  cplx r = v4;
  r = (m == 3) ? v3 : r;
  r = (m == 2) ? v2 : r;
  r = (m == 1) ? v1 : r;
  r = (m == 0) ? v0 : r;
  return r;
}

// ---------------------------------------------------------------------------
// Kernel 1: one block of 1024 threads (= W). Gaussian smoothing, wavelength
// min/max reduction, and all interpolation into the nk[5][W] table.
// ---------------------------------------------------------------------------
__global__ __launch_bounds__(KW)
void prep_kernel(const float* __restrict__ ri, const float* __restrict__ ec,
                 const float* __restrict__ fixed_data,
                 const float* __restrict__ wavelengths,
                 cplx* __restrict__ nkTab) {
  __shared__ float s_ns[KNSMOOTH];
  __shared__ float s_ks[KNSMOOTH];
  __shared__ float red[KW];
  __shared__ float s_lo, s_hi;
  const int t = threadIdx.x;

  // Gaussian-smooth n/k (np.convolve 'valid', kernel flipped; kernel is symmetric anyway)
  if (t < KNSMOOTH) {
    float kern[KKSIZE];
    float wsum = 0.0f;
    #pragma unroll
    for (int i = 0; i < KKSIZE; ++i) {
      float x = -10.0f + (float)i * (20.0f / (float)(KKSIZE - 1));
      float k = 0.5f * expf(-x * x * (1.0f / 32.0f));   // AMP * exp(-x^2 / (2*SIGMA^2))
      kern[i] = k;
      wsum += k;
    }
    float winv = 1.0f / wsum;
    float an = 0.0f, ak = 0.0f;
    #pragma unroll
    for (int m = 0; m < KKSIZE; ++m) {
      float wgt = kern[KKSIZE - 1 - m] * winv;
      an = fmaf(ri[t + m], wgt, an);
      ak = fmaf(ec[t + m], wgt, ak);
    }
    s_ns[t] = an;
    s_ks[t] = ak;
  }

  // min/max of wavelengths (deterministic trees)
  const float wl = wavelengths[t];
  red[t] = wl; __syncthreads();
  for (int off = KW / 2; off > 0; off >>= 1) {
    if (t < off) red[t] = fminf(red[t], red[t + off]);
    __syncthreads();
  }
  if (t == 0) s_lo = red[0];
  __syncthreads();
  red[t] = wl; __syncthreads();
  for (int off = KW / 2; off > 0; off >>= 1) {
    if (t < off) red[t] = fmaxf(red[t], red[t + off]);
    __syncthreads();
  }
  if (t == 0) s_hi = red[0];
  __syncthreads();

  // Unknown material: uniform-grid interp over dyn_wl = linspace(lo, hi, 331)
  {
    float tt = (wl - s_lo) / (s_hi - s_lo) * (float)(KNSMOOTH - 1);
    tt = fminf(fmaxf(tt, 0.0f), (float)(KNSMOOTH - 1));
    int i0 = (int)tt;
    if (i0 > KNSMOOTH - 2) i0 = KNSMOOTH - 2;
    float f = tt - (float)i0;
    float nu = fmaf(f, s_ns[i0 + 1] - s_ns[i0], s_ns[i0]);
    float ku = fmaf(f, s_ks[i0 + 1] - s_ks[i0], s_ks[i0]);
    nkTab[KNFIXED * KW + t] = { nu, ku };
  }

  // Fixed materials: jnp.interp on each material's own (monotonic) grid
  for (int m = 0; m < KNFIXED; ++m) {
    const float* g  = fixed_data + (m * 3 + 0) * KMAXD;
    const float* nv = fixed_data + (m * 3 + 1) * KMAXD;
    const float* kv = fixed_data + (m * 3 + 2) * KMAXD;
    float n, k;
    if (wl <= g[0]) {
      n = nv[0]; k = kv[0];
    } else if (wl >= g[KMAXD - 1]) {
      n = nv[KMAXD - 1]; k = kv[KMAXD - 1];
    } else {
      int lo = 0, hi = KMAXD - 1;
      while (hi - lo > 1) {
        int mid = (lo + hi) >> 1;
        if (g[mid] <= wl) lo = mid; else hi = mid;
      }
      float f = (wl - g[lo]) / (g[lo + 1] - g[lo]);
      n = fmaf(f, nv[lo + 1] - nv[lo], nv[lo]);
      k = fmaf(f, kv[lo + 1] - kv[lo], kv[lo]);
    }
    nkTab[m * KW + t] = { n, k };
  }
}

// ---------------------------------------------------------------------------
// Kernel 2: 65536 threads, one (wavelength, angle) pair each. 40-layer complex
// 2x2 transfer-matrix chain; block-level deterministic sum of absorbance.
// Lanes of a wave share a wavelength -> nk table loads are wave-uniform.
// ---------------------------------------------------------------------------
__global__ __launch_bounds__(KBLOCK)
void tmm_kernel(const cplx* __restrict__ nkTab,
                const float* __restrict__ wavelengths,
                const float* __restrict__ angles,
                const int* __restrict__ matdist,
                const float* __restrict__ th_above,
                const float* __restrict__ th_below,
                const float* __restrict__ unk_th,
                const float* __restrict__ comp,
                float* __restrict__ partials) {
  __shared__ int   s_mat[KNINT];
  __shared__ float s_d[KNINT];
  __shared__ float red[KBLOCK];
  const int t = threadIdx.x;

  if (t < KNINT) {
    s_mat[t] = matdist[t + 1];           // interior layers are entries 1..40
    float dv;
    if (t < KLABOVE)       dv = th_above[t];
    else if (t == KLABOVE) dv = unk_th[0] * (1.0f / 1000.0f) * comp[0];
    else                   dv = th_below[t - KLABOVE - 1];
    s_d[t] = dv;
  }
  __syncthreads();

  const int gid = blockIdx.x * KBLOCK + t;
  const int w = gid >> 6;          // gid / 64
  const int a = gid & 63;

  const float lam = wavelengths[w];
  const float ang = angles[a];
  float sa, ca;
  sincosf(ang, &sa, &ca);

  const int mat0 = matdist[0];
  const int matS = matdist[KNMATD - 1];

  const cplx n0   = nkTab[mat0 * KW + w];
  const cplx kx   = cscale(n0, sa);
  const cplx eta0 = cscale(n0, ca);
  const float invwl = TWO_PI / lam;

  // Per-distinct-material eta = n*cos and 1/eta (5 complex sqrts total)
  cplx eta[KNMAT], ie[KNMAT];
  #pragma unroll
  for (int m = 0; m < KNMAT; ++m) {
    cplx nm = nkTab[m * KW + w];
    cplx q  = cdiv(kx, nm);
    cplx c2 = { 1.0f - (q.x * q.x - q.y * q.y), -2.0f * q.x * q.y };
    cplx cj = csqrtc(c2);
    eta[m] = cmul(nm, cj);
    ie[m]  = crecip(eta[m]);
  }

  cplx M00 = { 1.0f, 0.0f }, M01 = { 0.0f, 0.0f };
  cplx M10 = { 0.0f, 0.0f }, M11 = { 0.0f, 0.0f };

  for (int j = 0; j < KNINT; ++j) {
    const int m = s_mat[j];
    const cplx em  = sel5(eta[0], eta[1], eta[2], eta[3], eta[4], m);
    const cplx iem = sel5(ie[0],  ie[1],  ie[2],  ie[3],  ie[4],  m);

    const float sc = invwl * s_d[j];
    const cplx delta = { em.x * sc, em.y * sc };

    float sr, cr;
    sincosf(delta.x, &sr, &cr);
    const float ep = expf(delta.y);
    const float en = expf(-delta.y);
    const float ch = 0.5f * (ep + en);
    const float sh = 0.5f * (ep - en);
    // cos(z) = cos x cosh y - i sin x sinh y ; sin(z) = sin x cosh y + i cos x sinh y
    const cplx c  = { cr * ch, -sr * sh };
    const cplx s  = { sr * ch,  cr * sh };
    const cplx is = { -s.y, s.x };        // i*s
    const cplx m01 = cmul(is, iem);
    const cplx m10 = cmul(is, em);

    const cplx A00 = cmadd(M00, c,   cmul(M01, m10));
    const cplx A01 = cmadd(M00, m01, cmul(M01, c));
    const cplx A10 = cmadd(M10, c,   cmul(M11, m10));
    const cplx A11 = cmadd(M10, m01, cmul(M11, c));
    M00 = A00; M01 = A01; M10 = A10; M11 = A11;
  }

  const cplx etas = sel5(eta[0], eta[1], eta[2], eta[3], eta[4], matS);
  const cplx p1 = cmul(eta0, cadd(M00, cmul(etas, M01)));   // eta0*M00 + eta0*etas*M01
  const cplx p2 = cadd(M10, cmul(etas, M11));               // M10 + etas*M11
  const cplx denom = cadd(p1, p2);
  const cplx idn = crecip(denom);
  const cplx r  = cmul(csub(p1, p2), idn);
  const cplx tt = cmul(cscale(eta0, 2.0f), idn);
  const float R = fmaf(r.x, r.x, r.y * r.y);
  const float T = (etas.x / eta0.x) * fmaf(tt.x, tt.x, tt.y * tt.y);
  const float absorb = 1.0f - R - T;

  // Deterministic block tree reduction
  red[t] = absorb; __syncthreads();
  for (int off = KBLOCK / 2; off > 0; off >>= 1) {
    if (t < off) red[t] += red[t + off];
    __syncthreads();
  }
  if (t == 0) partials[blockIdx.x] = red[0];
}

// ---------------------------------------------------------------------------
// Kernel 3: one wave32. Sum the 256 block partials with 4 chained
// V_WMMA_F32_16X16X4_F32 ops (B = ones => D[m][n] = row-sum of A + C), then
// fold rows with one shfl_xor. Full f32 precision, fully deterministic.
// ---------------------------------------------------------------------------
typedef __attribute__((ext_vector_type(2))) float v2f;
typedef __attribute__((ext_vector_type(8))) float v8f;

__global__ __launch_bounds__(32)
void reduce_kernel(const float* __restrict__ partials, float* __restrict__ out) {
  const int lane = threadIdx.x;
#if defined(__gfx1250__) && __has_builtin(__builtin_amdgcn_wmma_f32_16x16x4_f32)
  // A-matrix 16x4 f32 layout (ISA 7.12.2): lanes 0-15 hold K=0 (v0) / K=1 (v1)
  // for M = lane; lanes 16-31 hold K=2 / K=3 for M = lane-16.
  const int m4 = (lane & 15) * 4;
  const int kb = (lane < 16) ? 0 : 2;
  v2f B; B.x = 1.0f; B.y = 1.0f;
  v8f acc = {};
  #pragma unroll
  for (int c = 0; c < 4; ++c) {
    v2f Av;
    Av.x = partials[c * 64 + m4 + kb];
    Av.y = partials[c * 64 + m4 + kb + 1];
    acc = __builtin_amdgcn_wmma_f32_16x16x4_f32(
        /*neg_a=*/false, Av, /*neg_b=*/false, B,
        /*c_mod=*/(short)0, acc, /*reuse_a=*/false, /*reuse_b=*/false);
  }
  // acc[v] on lanes 0-15 holds row-sum for M=v; lanes 16-31 for M=8+v.
  float s = 0.0f;
  #pragma unroll
  for (int i = 0; i < 8; ++i) s += acc[i];
  const float other = __shfl_xor(s, 16, 32);
  const float total = s + other;
  if (lane == 0) out[0] = total * (1.0f / (float)(KW * KA));
#else
  if (lane == 0) {
    float s = 0.0f;
    for (int i = 0; i < KGRID; ++i) s += partials[i];
    out[0] = s * (1.0f / (float)(KW * KA));
  }
#endif
}

// ---------------------------------------------------------------------------
extern "C" void kernel_launch(void* const* d_in, const int* in_sizes, int n_in,
                              void* d_out, int out_size, void* d_ws, size_t ws_size,
                              hipStream_t stream) {
  const float* unk   = (const float*)d_in[0];   // unknown_layer_thickness (1)
  const float* ri    = (const float*)d_in[1];   // refractive_index (350)
  const float* ec    = (const float*)d_in[2];   // extinction_coefficient (350)
  const float* comp  = (const float*)d_in[3];   // compression_ratio (1)
  const float* fixd  = (const float*)d_in[4];   // fixed_data (4,3,400)
  const float* tha   = (const float*)d_in[5];   // thickness_above (20)
  const float* thb   = (const float*)d_in[6];   // thickness_below (19)
  const float* wls   = (const float*)d_in[7];   // wavelengths (1024)
  const float* angs  = (const float*)d_in[8];   // angles (64)
  const int*   mdist = (const int*)d_in[9];     // material_distribution (42)

  cplx*  nkTab    = (cplx*)d_ws;                                   // 5*1024*8B = 40 KB
  float* partials = (float*)((char*)d_ws + KNMAT * KW * sizeof(cplx)); // 256 floats

  prep_kernel<<<1, KW, 0, stream>>>(ri, ec, fixd, wls, nkTab);
  tmm_kernel<<<KGRID, KBLOCK, 0, stream>>>(nkTab, wls, angs, mdist, tha, thb,
                                           unk, comp, partials);
  reduce_kernel<<<1, 32, 0, stream>>>(partials, (float*)d_out);
}